// Transform_2327872274833
// MI455X (gfx1250) — compile-verified
//
#include <hip/hip_runtime.h>
#include <cstdint>

// ---------------------------------------------------------------------------
// Problem constants (match reference)
// ---------------------------------------------------------------------------
#define KB       8
#define KN       400000
#define KSCALE   20.0f
#define KFULL    4096.0f

#define P1_BLOCKS   64      // partial-reduction blocks per batch
#define P1_THREADS  256
#define P1_CHUNK    (KN / P1_BLOCKS)   // 6250

#define TILE        1024    // points per block in the elementwise pass
#define P3_THREADS  256     // 4 points per thread

// Workspace layout (floats): partials [B*P1_BLOCKS*8], then offsets [B*4]
#define WS_PART_FLOATS (KB * P1_BLOCKS * 8)

// ---------------------------------------------------------------------------
// Tensor Data Mover: 1-D contiguous dword tile, global -> LDS  (ISA ch.8 D#)
// ---------------------------------------------------------------------------
typedef unsigned int tdm_u32x4 __attribute__((ext_vector_type(4)));
typedef int          tdm_i32x8 __attribute__((ext_vector_type(8)));
typedef int          tdm_i32x4 __attribute__((ext_vector_type(4)));

__device__ __forceinline__ void tdm_load_dwords_to_lds(unsigned lds_byte_off,
                                                       const void* gsrc,
                                                       unsigned ndwords) {
  unsigned long long ga = (unsigned long long)(uintptr_t)gsrc;

  // D# group 0: count=1 | lds_addr | global_addr[56:0] | type=2
  tdm_u32x4 g0;
  g0[0] = 1u;
  g0[1] = lds_byte_off;
  g0[2] = (unsigned)(ga & 0xffffffffull);
  g0[3] = (unsigned)((ga >> 32) & 0x01ffffffull) | (2u << 30);

  // D# group 1: data_size=4B, tensor_dim0=ndwords, tensor_dim1=1,
  //             tile_dim0=ndwords, tile_dim1=1, stride0=ndwords
  tdm_i32x8 g1;
  g1[0] = (int)(2u << 16);                                   // data_size=2 (4B)
  g1[1] = (int)((ndwords & 0xffffu) << 16);                  // tensor_dim0[15:0]
  g1[2] = (int)(((ndwords >> 16) & 0xffffu) | (1u << 16));   // tdim0[31:16] | tdim1=1
  g1[3] = (int)((ndwords & 0xffffu) << 16);                  // tile_dim0
  g1[4] = 1;                                                 // tile_dim1=1, tile_dim2=0
  g1[5] = (int)ndwords;                                      // tensor_dim0_stride[31:0]
  g1[6] = 0;
  g1[7] = 0;

  tdm_i32x4 z4 = {0, 0, 0, 0};
#if defined(__clang_major__) && (__clang_major__ >= 23)
  tdm_i32x8 z8 = {0, 0, 0, 0, 0, 0, 0, 0};
  __builtin_amdgcn_tensor_load_to_lds(g0, g1, z4, z4, z8, 0);
#else
  __builtin_amdgcn_tensor_load_to_lds(g0, g1, z4, z4, 0);
#endif
}

__device__ __forceinline__ unsigned lds_off(const void* p) {
  // Generic pointers to LDS carry the workgroup-relative byte offset in the
  // low 32 bits (aperture in the high 32) -- ISA 10.2 aperture mapping.
  return (unsigned)(uintptr_t)p;
}

// ---------------------------------------------------------------------------
// Pass 1: per-(batch, block) partial min/max of 20*xyz over a chunk of points
// ---------------------------------------------------------------------------
__global__ void __launch_bounds__(P1_THREADS)
k_minmax_partial(const float* __restrict__ points, float* __restrict__ ws_part) {
  const int b = blockIdx.x / P1_BLOCKS;
  const int j = blockIdx.x % P1_BLOCKS;
  const int start = j * P1_CHUNK;
  const float4* pb = (const float4*)points + (size_t)b * KN;

  float mnx =  __builtin_inff(), mny =  __builtin_inff(), mnz =  __builtin_inff();
  float mxx = -__builtin_inff(), mxy = -__builtin_inff(), mxz = -__builtin_inff();

  for (int i = start + (int)threadIdx.x; i < start + P1_CHUNK; i += P1_THREADS) {
    float4 p = pb[i];                          // global_load_b128
    float ax = __fmul_rn(KSCALE, p.x);
    float ay = __fmul_rn(KSCALE, p.y);
    float az = __fmul_rn(KSCALE, p.z);
    mnx = fminf(mnx, ax); mxx = fmaxf(mxx, ax);
    mny = fminf(mny, ay); mxy = fmaxf(mxy, ay);
    mnz = fminf(mnz, az); mxz = fmaxf(mxz, az);
  }

  // wave32 butterfly reduction
  for (int m = 16; m >= 1; m >>= 1) {
    mnx = fminf(mnx, __shfl_xor(mnx, m, 32));
    mny = fminf(mny, __shfl_xor(mny, m, 32));
    mnz = fminf(mnz, __shfl_xor(mnz, m, 32));
    mxx = fmaxf(mxx, __shfl_xor(mxx, m, 32));
    mxy = fmaxf(mxy, __shfl_xor(mxy, m, 32));
    mxz = fmaxf(mxz, __shfl_xor(mxz, m, 32));
  }

  __shared__ float wred[P1_THREADS / 32][8];
  const int wave = threadIdx.x >> 5;
  if ((threadIdx.x & 31) == 0) {
    wred[wave][0] = mnx; wred[wave][1] = mny; wred[wave][2] = mnz;
    wred[wave][3] = mxx; wred[wave][4] = mxy; wred[wave][5] = mxz;
  }
  __syncthreads();
  if (threadIdx.x == 0) {
    for (int w = 1; w < P1_THREADS / 32; ++w) {
      wred[0][0] = fminf(wred[0][0], wred[w][0]);
      wred[0][1] = fminf(wred[0][1], wred[w][1]);
      wred[0][2] = fminf(wred[0][2], wred[w][2]);
      wred[0][3] = fmaxf(wred[0][3], wred[w][3]);
      wred[0][4] = fmaxf(wred[0][4], wred[w][4]);
      wred[0][5] = fmaxf(wred[0][5], wred[w][5]);
    }
    float* o = ws_part + (size_t)(b * P1_BLOCKS + j) * 8;
    o[0] = wred[0][0]; o[1] = wred[0][1]; o[2] = wred[0][2];
    o[3] = wred[0][3]; o[4] = wred[0][4]; o[5] = wred[0][5];
  }
}

// ---------------------------------------------------------------------------
// Pass 2: reduce partials per batch, compute placement offset [B,3]
// ---------------------------------------------------------------------------
__global__ void __launch_bounds__(64)
k_offset(const float* __restrict__ ws_part, const float* __restrict__ rand_u,
         float* __restrict__ ws_off, float* __restrict__ out_off) {
  const int b = blockIdx.x;
  const int t = threadIdx.x;

  __shared__ float red[P1_BLOCKS][8];
  const float* p = ws_part + (size_t)(b * P1_BLOCKS + t) * 8;
  red[t][0] = p[0]; red[t][1] = p[1]; red[t][2] = p[2];
  red[t][3] = p[3]; red[t][4] = p[4]; red[t][5] = p[5];
  __syncthreads();

  for (int s = P1_BLOCKS / 2; s >= 1; s >>= 1) {
    if (t < s) {
      red[t][0] = fminf(red[t][0], red[t + s][0]);
      red[t][1] = fminf(red[t][1], red[t + s][1]);
      red[t][2] = fminf(red[t][2], red[t + s][2]);
      red[t][3] = fmaxf(red[t][3], red[t + s][3]);
      red[t][4] = fmaxf(red[t][4], red[t + s][4]);
      red[t][5] = fmaxf(red[t][5], red[t + s][5]);
    }
    __syncthreads();
  }

  if (t < 3) {
    const float m = red[0][t];
    const float M = red[0][3 + t];
    const float room = KFULL - (M - m);
    const float u1 = rand_u[b * 6 + t];        // rand_u[b,0,t]
    const float u2 = rand_u[b * 6 + 3 + t];    // rand_u[b,1,t]
    const float off = -m + fmaxf(room - 1e-3f, 0.0f) * u1
                         + fminf(room + 1e-3f, 0.0f) * u2;
    ws_off[b * 4 + t]  = off;   // padded broadcast for pass 3 (float4 load)
    out_off[b * 3 + t] = off;   // 4th output
  }
}

// ---------------------------------------------------------------------------
// Pass 3: TDM-staged elementwise transform. 1 block = 1024 points (32 KB LDS).
// ---------------------------------------------------------------------------
__global__ void __launch_bounds__(P3_THREADS)
k_transform(const float* __restrict__ points, const float* __restrict__ feas,
            const int* __restrict__ masks, const float* __restrict__ noise,
            const float* __restrict__ ws_off, float* __restrict__ out) {
  __shared__ float s_pts[TILE * 4];   // 16 KB
  __shared__ float s_fea[TILE * 3];   // 12 KB
  __shared__ int   s_msk[TILE];       //  4 KB

  const int tileStart = (int)blockIdx.x * TILE;   // total points = 3.2M < 2^31

  // Wave 0 issues three TDM DMA descriptors, waits on TENSORcnt, then the
  // workgroup barrier publishes the LDS tile to all 8 waves.
  if (threadIdx.x < 32) {
    tdm_load_dwords_to_lds(lds_off(s_pts), points + (size_t)tileStart * 4, TILE * 4);
    tdm_load_dwords_to_lds(lds_off(s_fea), feas   + (size_t)tileStart * 3, TILE * 3);
    tdm_load_dwords_to_lds(lds_off(s_msk), masks  + (size_t)tileStart,     TILE);
    __builtin_amdgcn_s_wait_tensorcnt(0);
  }
  __syncthreads();

  const int l0 = (int)threadIdx.x * 4;     // 4 points/thread; N%4==0 keeps all
  const int g0 = tileStart + l0;           // 4 points in the same batch
  const int b  = g0 / KN;

  const float4 offv = *(const float4*)(ws_off + b * 4);
  const float nzx = noise[b * 3 + 0];
  const float nzy = noise[b * 3 + 1];
  const float nzz = noise[b * 3 + 2];

  const float4* sp  = (const float4*)s_pts;
  const float4* sf  = (const float4*)(s_fea + (size_t)l0 * 3);  // 48B-aligned
  const int4    mk4 = *(const int4*)(s_msk + l0);

  const float4 f4a = sf[0], f4b = sf[1], f4c = sf[2];
  const float fe[12] = { f4a.x, f4a.y, f4a.z, f4a.w,
                         f4b.x, f4b.y, f4b.z, f4b.w,
                         f4c.x, f4c.y, f4c.z, f4c.w };
  const int mks[4] = { mk4.x, mk4.y, mk4.z, mk4.w };

  float4 locs[4];
  float  fo[12];
  float  mo[4];

  #pragma unroll
  for (int k = 0; k < 4; ++k) {
    const float4 p = sp[l0 + k];
    // two-rounding 20*p + off (matches reference; no FMA contraction)
    const float ax = __fadd_rn(__fmul_rn(KSCALE, p.x), offv.x);
    const float ay = __fadd_rn(__fmul_rn(KSCALE, p.y), offv.y);
    const float az = __fadd_rn(__fmul_rn(KSCALE, p.z), offv.z);
    const bool inbox = (ax >= 0.0f) && (ay >= 0.0f) && (az >= 0.0f) &&
                       (ax < KFULL) && (ay < KFULL) && (az < KFULL);
    const int  mk    = mks[k];
    const bool valid = inbox && (mk > 0);

    locs[k].x = valid ? (float)(int)ax : 0.0f;
    locs[k].y = valid ? (float)(int)ay : 0.0f;
    locs[k].z = valid ? (float)(int)az : 0.0f;
    locs[k].w = valid ? (float)b       : 0.0f;

    fo[k * 3 + 0] = valid ? __fadd_rn(fe[k * 3 + 0], nzx) : 0.0f;
    fo[k * 3 + 1] = valid ? __fadd_rn(fe[k * 3 + 1], nzy) : 0.0f;
    fo[k * 3 + 2] = valid ? __fadd_rn(fe[k * 3 + 2], nzz) : 0.0f;

    mo[k] = valid ? (float)mk : 0.0f;
  }

  // locs: [B,N,4] at out[0)
  float4* outL = (float4*)out;
  #pragma unroll
  for (int k = 0; k < 4; ++k) outL[(size_t)g0 + k] = locs[k];

  // feats: [B,N,3] at out[B*N*4)
  float* outF = out + (size_t)KB * KN * 4;
  float4* pf = (float4*)(outF + (size_t)g0 * 3);                // 48B-aligned
  pf[0] = make_float4(fo[0], fo[1], fo[2],  fo[3]);
  pf[1] = make_float4(fo[4], fo[5], fo[6],  fo[7]);
  pf[2] = make_float4(fo[8], fo[9], fo[10], fo[11]);

  // new_masks: [B*N] at out[B*N*7)
  float* outM = out + (size_t)KB * KN * 7;
  *(float4*)(outM + g0) = make_float4(mo[0], mo[1], mo[2], mo[3]);
}

// ---------------------------------------------------------------------------
// Launch
// ---------------------------------------------------------------------------
extern "C" void kernel_launch(void* const* d_in, const int* in_sizes, int n_in,
                              void* d_out, int out_size, void* d_ws, size_t ws_size,
                              hipStream_t stream) {
  (void)in_sizes; (void)n_in; (void)out_size; (void)ws_size;

  const float* points = (const float*)d_in[0];   // [B,N,4]
  const float* feas   = (const float*)d_in[1];   // [B,N,3]
  const int*   masks  = (const int*)  d_in[2];   // [B*N]
  const float* rand_u = (const float*)d_in[3];   // [B,2,3]
  const float* noise  = (const float*)d_in[4];   // [B,3]

  float* out     = (float*)d_out;
  float* ws      = (float*)d_ws;
  float* ws_part = ws;                           // B*P1_BLOCKS*8 floats
  float* ws_off  = ws + WS_PART_FLOATS;          // B*4 floats
  float* out_off = out + (size_t)KB * KN * 8;    // offset output section

  k_minmax_partial<<<dim3(KB * P1_BLOCKS), dim3(P1_THREADS), 0, stream>>>(points, ws_part);
  k_offset<<<dim3(KB), dim3(P1_BLOCKS), 0, stream>>>(ws_part, rand_u, ws_off, out_off);
  k_transform<<<dim3((KB * KN) / TILE), dim3(P3_THREADS), 0, stream>>>(
      points, feas, masks, noise, ws_off, out);
}